// SelfAttention_82016695485289
// MI455X (gfx1250) — compile-verified
//
#include <hip/hip_runtime.h>
#include <math.h>

#define B_  16
#define C_  64
#define H_  256
#define W_  256
#define HW  (H_ * W_)

typedef __attribute__((ext_vector_type(2))) float v2f;
typedef __attribute__((ext_vector_type(8))) float v8f;
typedef __attribute__((ext_vector_type(4))) int   v4i;

// D = A(16x4, f32) * B(4x16, f32) + C(16x16, f32)   -> v_wmma_f32_16x16x4_f32
__device__ __forceinline__ v8f wmma4(v2f a, v2f b, v8f c) {
    return __builtin_amdgcn_wmma_f32_16x16x4_f32(
        /*neg_a=*/false, a, /*neg_b=*/false, b,
        /*c_mod=*/(short)0, c, /*reuse_a=*/false, /*reuse_b=*/false);
}

// 8-byte-aligned LDS fragment load -> single ds_load_b64 into an aligned pair
__device__ __forceinline__ v2f ldsv2(const float* p) {
    return *(const v2f*)__builtin_assume_aligned(p, 8);
}

#define PAD 4

// ---- async global -> LDS (CDNA5 path; falls back to float4 copy) ----------
#if defined(__gfx1250__) && __has_builtin(__builtin_amdgcn_global_load_async_to_lds_b128)
#define HAS_ASYNC_LDS 1
#else
#define HAS_ASYNC_LDS 0
#endif

__device__ __forceinline__ void copy16_to_lds(float* lds, const float* g) {
#if HAS_ASYNC_LDS
    __builtin_amdgcn_global_load_async_to_lds_b128(
        (__attribute__((address_space(1))) v4i*)g,      // global src (16B)
        (__attribute__((address_space(3))) v4i*)lds,    // LDS dst
        /*imm offset=*/0, /*cpol=*/0);
#else
    *(float4*)lds = *(const float4*)g;
#endif
}

__device__ __forceinline__ void tile_fill_fence() {
#if HAS_ASYNC_LDS
#if __has_builtin(__builtin_amdgcn_s_wait_asynccnt)
    __builtin_amdgcn_s_wait_asynccnt(0);
#else
    asm volatile("s_wait_asynccnt 0" ::: "memory");
#endif
#endif
    __syncthreads();
}

// ---------------------------------------------------------------------------
// 1x1 conv projection:  Out[b][o][p] = sum_c W[o][c] * X[b][c][p]
// Block: 256 pixels of one batch; K = 64 entirely in LDS.
// Each wave: 16 output channels x 128 pixels (8 tiles sharing the A fragment).
// ---------------------------------------------------------------------------
__global__ __launch_bounds__(256) void proj_kernel(const float* __restrict__ X,
                                                   const float* __restrict__ Wm,
                                                   float* __restrict__ Out) {
    __shared__ float Ws[C_][C_ + PAD];        // 64 x 68   (~17.4 KB)
    __shared__ float Xs[C_][256 + PAD];       // 64 x 260  (~66.6 KB)

    const int tid  = threadIdx.x;
    const int b    = blockIdx.y;
    const int pix0 = blockIdx.x * 256;

    for (int i = tid; i < C_ * 16; i += 256) {            // 16B chunks of W
        int o = i >> 4, c4 = (i & 15) * 4;
        copy16_to_lds(&Ws[o][c4], Wm + o * C_ + c4);
    }
    const float* xb = X + (size_t)b * C_ * HW + pix0;
    for (int i = tid; i < C_ * 64; i += 256) {            // 16B chunks of X
        int c = i >> 6, p4 = (i & 63) * 4;
        copy16_to_lds(&Xs[c][p4], xb + (size_t)c * HW + p4);
    }
    tile_fill_fence();

    const int lane  = tid & 31;
    const int wv    = tid >> 5;
    const int oRow  = (wv & 3) * 16;                      // output-channel tile
    const int cHalf = (wv >> 2) * 128;                    // pixel half
    const int n     = lane & 15;
    const int kh    = (lane >> 4) * 2;
    const int mB    = (lane >> 4) * 8;

    v8f acc[8];
#pragma unroll
    for (int t = 0; t < 8; ++t) acc[t] = (v8f){};

#pragma unroll
    for (int kk = 0; kk < C_; kk += 4) {
        const v2f a = ldsv2(&Ws[oRow + n][kk + kh]);
#pragma unroll
        for (int t = 0; t < 8; ++t) {
            // B[k][p]: two K-rows of X at the same pixel column -> not
            // contiguous, keep element form (column access).
            const int col0 = cHalf + t * 16;
            v2f bf;
            bf.x = Xs[kk + kh][col0 + n];
            bf.y = Xs[kk + kh + 1][col0 + n];
            acc[t] = wmma4(a, bf, acc[t]);
        }
    }

    float* outb = Out + (size_t)b * C_ * HW + pix0;
#pragma unroll
    for (int t = 0; t < 8; ++t) {
        const int col0 = cHalf + t * 16;
#pragma unroll
        for (int r = 0; r < 8; ++r)
            outb[(size_t)(oRow + mB + r) * HW + col0 + n] = acc[t][r];
    }
}

// ---------------------------------------------------------------------------
// scores[b,c] = (Q[b,c] * K[b,c]^T) / 8    (256x256x256 fp32 GEMM per (b,c))
// 128x128 output tile per workgroup; whole K=256 contraction in LDS (266 KB).
// Each wave: one 16x128 strip (8 tiles share the A fragment each K-step).
// Both A and B fragments are contiguous 8B-aligned LDS pairs -> ds_load_b64.
// ---------------------------------------------------------------------------
__global__ __launch_bounds__(256) void scores_kernel(const float* __restrict__ Q,
                                                     const float* __restrict__ K,
                                                     float* __restrict__ S) {
    __shared__ float Qs[128][256 + PAD];      // 133 KB
    __shared__ float Ks[128][256 + PAD];      // 133 KB

    const int tid = threadIdx.x;
    const int bc  = blockIdx.z;               // b*C + c
    const int i0  = blockIdx.x * 128;         // Q row block
    const int j0  = blockIdx.y * 128;         // K row block

    const float* qm = Q + (size_t)bc * HW;
    const float* km = K + (size_t)bc * HW;
    for (int i = tid; i < 128 * 64; i += 256) {           // 16B chunks
        int r = i >> 6, w4 = (i & 63) * 4;
        copy16_to_lds(&Qs[r][w4], qm + (size_t)(i0 + r) * W_ + w4);
        copy16_to_lds(&Ks[r][w4], km + (size_t)(j0 + r) * W_ + w4);
    }
    tile_fill_fence();

    const int lane = tid & 31;
    const int wv   = tid >> 5;
    const int n    = lane & 15;
    const int kh   = (lane >> 4) * 2;
    const int mB   = (lane >> 4) * 8;
    const int r0   = wv * 16;                 // wave's row strip within 128

    v8f acc[8];
#pragma unroll
    for (int t = 0; t < 8; ++t) acc[t] = (v8f){};

#pragma unroll 4
    for (int kk = 0; kk < 256; kk += 4) {
        const v2f a = ldsv2(&Qs[r0 + n][kk + kh]);        // A[m][k]
#pragma unroll
        for (int t = 0; t < 8; ++t) {
            const v2f bf = ldsv2(&Ks[t * 16 + n][kk + kh]); // B[k][n]=K^T
            acc[t] = wmma4(a, bf, acc[t]);
        }
    }

    float* sm = S + (size_t)bc * H_ * H_;
#pragma unroll
    for (int t = 0; t < 8; ++t)
#pragma unroll
        for (int r = 0; r < 8; ++r)
            sm[(size_t)(i0 + r0 + mB + r) * H_ + j0 + t * 16 + n] = acc[t][r] * 0.125f;
}

// ---------------------------------------------------------------------------
// softmax over the CHANNEL axis (dim=1): for each (b,i,j), across 64 channels
// at stride H*H. All 64 values held in registers; in-place on scores.
// ---------------------------------------------------------------------------
__global__ __launch_bounds__(256) void softmax_c_kernel(float* __restrict__ S) {
    const size_t gid = (size_t)blockIdx.x * 256 + threadIdx.x;    // over B*H*H
    const size_t b   = gid / (size_t)(H_ * H_);
    const size_t rem = gid % (size_t)(H_ * H_);
    float* base = S + b * (size_t)C_ * H_ * H_ + rem;

    float v[C_];
    float mx = -3.402823466e38f;
#pragma unroll
    for (int c = 0; c < C_; ++c) {
        v[c] = base[(size_t)c * H_ * H_];
        mx = fmaxf(mx, v[c]);
    }
    float s = 0.0f;
#pragma unroll
    for (int c = 0; c < C_; ++c) {
        v[c] = __expf(v[c] - mx);
        s += v[c];
    }
    const float inv = 1.0f / s;
#pragma unroll
    for (int c = 0; c < C_; ++c)
        base[(size_t)c * H_ * H_] = v[c] * inv;
}

// ---------------------------------------------------------------------------
// out[b,c] = attn[b,c] (256x256) * V[b,c] (256x256)
// 128x128 output tile; attn rows + V column panel resident in LDS (268 KB).
// ---------------------------------------------------------------------------
__global__ __launch_bounds__(256) void attnv_kernel(const float* __restrict__ A,
                                                    const float* __restrict__ V,
                                                    float* __restrict__ O) {
    __shared__ float As[128][256 + PAD];      // 133 KB (attn rows, all g)
    __shared__ float Vs[256][128 + PAD];      // 135 KB (all g, 128 w-cols)

    const int tid = threadIdx.x;
    const int bc  = blockIdx.z;
    const int i0  = blockIdx.x * 128;         // attn row block (h)
    const int w0  = blockIdx.y * 128;         // V col block (w)

    const float* am = A + (size_t)bc * H_ * H_;
    const float* vm = V + (size_t)bc * HW;
    for (int i = tid; i < 128 * 64; i += 256) {           // attn: 16B chunks
        int r = i >> 6, g4 = (i & 63) * 4;
        copy16_to_lds(&As[r][g4], am + (size_t)(i0 + r) * H_ + g4);
    }
    for (int i = tid; i < 256 * 32; i += 256) {           // V: 16B chunks
        int g = i >> 5, w4 = (i & 31) * 4;
        copy16_to_lds(&Vs[g][w4], vm + (size_t)g * W_ + w0 + w4);
    }
    tile_fill_fence();

    const int lane = tid & 31;
    const int wv   = tid >> 5;
    const int n    = lane & 15;
    const int kh   = (lane >> 4) * 2;
    const int mB   = (lane >> 4) * 8;
    const int r0   = wv * 16;

    v8f acc[8];
#pragma unroll
    for (int t = 0; t < 8; ++t) acc[t] = (v8f){};

#pragma unroll 4
    for (int kk = 0; kk < 256; kk += 4) {
        const v2f a = ldsv2(&As[r0 + n][kk + kh]);        // A[m][g] contiguous
#pragma unroll
        for (int t = 0; t < 8; ++t) {
            // B[g][w]: two g-rows of V at same w column -> column access.
            v2f bf;
            bf.x = Vs[kk + kh][t * 16 + n];
            bf.y = Vs[kk + kh + 1][t * 16 + n];
            acc[t] = wmma4(a, bf, acc[t]);
        }
    }

    float* om = O + (size_t)bc * HW;
#pragma unroll
    for (int t = 0; t < 8; ++t)
#pragma unroll
        for (int r = 0; r < 8; ++r)
            om[(size_t)(i0 + r0 + mB + r) * W_ + w0 + t * 16 + n] = acc[t][r];
}

// ---------------------------------------------------------------------------
extern "C" void kernel_launch(void* const* d_in, const int* in_sizes, int n_in,
                              void* d_out, int out_size, void* d_ws, size_t ws_size,
                              hipStream_t stream) {
    (void)in_sizes; (void)n_in; (void)out_size; (void)ws_size;

    const float* query  = (const float*)d_in[0];
    const float* keys   = (const float*)d_in[1];
    const float* values = (const float*)d_in[2];
    const float* Wq     = (const float*)d_in[3];
    const float* Wk     = (const float*)d_in[4];
    const float* Wv     = (const float*)d_in[5];
    float* out = (float*)d_out;
    float* ws  = (float*)d_ws;

    const size_t T = (size_t)B_ * C_ * HW;    // 67,108,864 floats (256 MB)
    float* qp = ws;                           // projected Q
    float* kp = ws + T;                       // projected K
    float* vp = ws + 2 * T;                   // projected V
    float* sc = ws + 3 * T;                   // scores / attn (in-place softmax)

    dim3 pg(HW / 256, B_);
    proj_kernel<<<pg, 256, 0, stream>>>(query,  Wq, qp);
    proj_kernel<<<pg, 256, 0, stream>>>(keys,   Wk, kp);
    proj_kernel<<<pg, 256, 0, stream>>>(values, Wv, vp);

    dim3 sg(H_ / 128, H_ / 128, B_ * C_);
    scores_kernel<<<sg, 256, 0, stream>>>(qp, kp, sc);

    softmax_c_kernel<<<(B_ * H_ * H_) / 256, 256, 0, stream>>>(sc);

    dim3 og(H_ / 128, W_ / 128, B_ * C_);
    attnv_kernel<<<og, 256, 0, stream>>>(sc, vp, out);
}